// YoloLoss_85856396247774
// MI455X (gfx1250) — compile-verified
//
#include <hip/hip_runtime.h>

// ---------------------------------------------------------------------------
// YOLO-v1 loss, MI455X (gfx1250, wave32).
// Bandwidth-bound streaming reduction (~174 MB -> ~7.5us at 23.3 TB/s).
// Kernel 1: grid-stride pass, register accumulation of 4 loss terms,
//           wave reduction via V_WMMA_F32_16X16X4_F32 (2 accumulators per op),
//           cross-wave reduction in LDS, per-block partials to d_ws.
// Kernel 2: deterministic final reduction + scaling of the 5 outputs.
// ---------------------------------------------------------------------------

#define S_GRID   14
#define N_BATCH  4096
#define NCELL    (N_BATCH * S_GRID * S_GRID)   // 802816
#define NUM_CLS  20
#define NBLOCKS  1024
#define NTHREADS 256

typedef float v2f __attribute__((ext_vector_type(2)));
typedef float v8f __attribute__((ext_vector_type(8)));

// Reduce TWO per-lane accumulators across a full wave32 with one WMMA.
// A (16x4 f32): VGPR0 = a0, VGPR1 = a1  ->  row m = {a0_m, a1_m, a0_{16+m}, a1_{16+m}}
// B (4x16 f32): selector, 1.0 for rows {0,2} on cols 0..7 and rows {1,3} on cols 8..15.
// D[m][n<8]  = a0_m + a0_{16+m};  D[m][n>=8] = a1_m + a1_{16+m}.
// Summing D's 8 VGPRs in-lane + shfl_xor(16) gives full-wave sums:
// columns 0..7 hold sum(a0), columns 8..15 hold sum(a1).
__device__ __forceinline__ void wave_reduce_pair(float a0, float a1,
                                                 float& s0, float& s1) {
#if __has_builtin(__builtin_amdgcn_wmma_f32_16x16x4_f32)
  const int lane = threadIdx.x & 31;
  // bsel = 1 on lanes [0..7] and [24..31]  (bit3 == bit4)
  const float bsel = (((lane >> 3) & 1) == ((lane >> 4) & 1)) ? 1.0f : 0.0f;
  v2f A; A[0] = a0;   A[1] = a1;
  v2f B; B[0] = bsel; B[1] = bsel;
  v8f C = {};
  v8f D = __builtin_amdgcn_wmma_f32_16x16x4_f32(
      /*neg_a=*/false, A, /*neg_b=*/false, B,
      /*c_mod=*/(short)0, C, /*reuse_a=*/false, /*reuse_b=*/false);
  float s = D[0] + D[1] + D[2] + D[3] + D[4] + D[5] + D[6] + D[7];
  s += __shfl_xor(s, 16, 32);
  s0 = __shfl(s, 0, 32);   // column 0  -> total of a0
  s1 = __shfl(s, 8, 32);   // column 8  -> total of a1
#else
  float t0 = a0, t1 = a1;
  for (int m = 1; m < 32; m <<= 1) {
    t0 += __shfl_xor(t0, m, 32);
    t1 += __shfl_xor(t1, m, 32);
  }
  s0 = t0; s1 = t1;
#endif
}

__global__ void __launch_bounds__(NTHREADS)
yolo_partial_kernel(const float* __restrict__ pred,          // [N,S,S,30]
                    const float* __restrict__ tbox,          // [N,S,S,4]
                    const float* __restrict__ tcls,          // [N,S,S,20]
                    const unsigned char* __restrict__ mask,  // [N,S,S] bool
                    float* __restrict__ partial) {           // [NBLOCKS*4]
  float acc_reg = 0.0f, acc_con = 0.0f, acc_no = 0.0f, acc_cls = 0.0f;

  const int gid    = blockIdx.x * blockDim.x + threadIdx.x;
  const int stride = gridDim.x * blockDim.x;

  for (int cell = gid; cell < NCELL; cell += stride) {
    const float maskf = mask[cell] ? 1.0f : 0.0f;

    // pred row: 30 floats, 120-byte pitch -> 8-byte aligned -> float2 loads
    const float2* p2 = (const float2*)(pred + (size_t)cell * 30);
    const float2 q0 = p2[0], q1 = p2[1], q2 = p2[2], q3 = p2[3], q4 = p2[4];

    // ---- class loss: sum_c (pred_cls - target_cls)^2 * mask ----
    const float2* pc2 = p2 + 5;                                   // p[10..29]
    const float2* tc2 = (const float2*)(tcls + (size_t)cell * 20);
    float cls = 0.0f;
#pragma unroll
    for (int i = 0; i < NUM_CLS / 2; ++i) {
      const float2 a = pc2[i], b = tc2[i];
      const float dx = a.x - b.x, dy = a.y - b.y;
      cls = fmaf(dx, dx, fmaf(dy, dy, cls));
    }
    acc_cls += maskf * cls;

    // ---- no-object confidence loss (conf at offsets 4 and 9) ----
    const float c0 = q2.x, c1 = q4.y;
    acc_no += (1.0f - maskf) * (c0 * c0 + c1 * c1);

    // ---- target box -> xyxy ----
    const float4 tb = ((const float4*)tbox)[cell];                // 16B aligned
    const float gcx = tb.x / 14.0f, gcy = tb.y / 14.0f;
    const float gx1 = gcx - 0.5f * tb.z, gy1 = gcy - 0.5f * tb.w;
    const float gx2 = gcx + 0.5f * tb.z, gy2 = gcy + 0.5f * tb.w;
    const float area_g = (gx2 - gx1) * (gy2 - gy1);

    // ---- IoU for the two predicted boxes ----
    const float bx[2] = {q0.x, q2.y};
    const float by[2] = {q0.y, q3.x};
    const float bw[2] = {q1.x, q3.y};
    const float bh[2] = {q1.y, q4.x};
    float iou[2];
#pragma unroll
    for (int b = 0; b < 2; ++b) {
      const float cx = bx[b] / 14.0f, cy = by[b] / 14.0f;
      const float px1 = cx - 0.5f * bw[b], py1 = cy - 0.5f * bh[b];
      const float px2 = cx + 0.5f * bw[b], py2 = cy + 0.5f * bh[b];
      const float ltx = fmaxf(px1, gx1), lty = fmaxf(py1, gy1);
      const float rbx = fminf(px2, gx2), rby = fminf(py2, gy2);
      const float iw = fmaxf(rbx - ltx, 0.0f), ih = fmaxf(rby - lty, 0.0f);
      const float inter  = iw * ih;
      const float area_p = (px2 - px1) * (py2 - py1);
      iou[b] = inter / (area_p + area_g - inter);
    }
    // argmax with first-index tie-break -> strict greater-than
    const int best = (iou[1] > iou[0]) ? 1 : 0;
    const float sx = bx[best], sy = by[best];
    const float sw = bw[best], sh = bh[best];
    const float sc = best ? c1 : c0;

    // ---- regression + containing-confidence ----
    const float dx = sx - tb.x, dy = sy - tb.y;
    const float dw = sqrtf(sw) - sqrtf(tb.z);
    const float dh = sqrtf(sh) - sqrtf(tb.w);
    acc_reg += maskf * (dx * dx + dy * dy + dw * dw + dh * dh);
    const float dc = sc - 1.0f;
    acc_con += maskf * dc * dc;
  }

  // ---- wave reduction via WMMA (uniform control flow: EXEC all-ones) ----
  float s_reg, s_con, s_no, s_cls;
  wave_reduce_pair(acc_reg, acc_con, s_reg, s_con);
  wave_reduce_pair(acc_no,  acc_cls, s_no,  s_cls);

  // ---- cross-wave reduction (8 waves / block) ----
  __shared__ float lds[NTHREADS / 32][4];
  const int lane = threadIdx.x & 31;
  const int wave = threadIdx.x >> 5;
  if (lane == 0) {
    lds[wave][0] = s_reg; lds[wave][1] = s_con;
    lds[wave][2] = s_no;  lds[wave][3] = s_cls;
  }
  __syncthreads();
  if (threadIdx.x < 32) {   // 8 waves x 4 terms = 32 values
    const int w = threadIdx.x >> 2, j = threadIdx.x & 3;
    float v = lds[w][j];
    v += __shfl_xor(v, 4, 32);
    v += __shfl_xor(v, 8, 32);
    v += __shfl_xor(v, 16, 32);
    if (threadIdx.x < 4) partial[blockIdx.x * 4 + j] = v;
  }
}

__global__ void __launch_bounds__(256)
yolo_final_kernel(const float* __restrict__ partial,  // [NBLOCKS*4]
                  float* __restrict__ out) {          // [5]
  const int t = threadIdx.x;       // 256 threads, single block
  const int j = t & 3;
  float acc = 0.0f;
  for (int i = (t >> 2); i < NBLOCKS; i += 64) acc += partial[i * 4 + j];
  acc += __shfl_xor(acc, 4, 32);
  acc += __shfl_xor(acc, 8, 32);
  acc += __shfl_xor(acc, 16, 32);

  __shared__ float lds[8][4];
  const int lane = t & 31, wave = t >> 5;
  if (lane < 4) lds[wave][lane] = acc;   // lane == j for lanes 0..3
  __syncthreads();

  if (t < 32) {
    const int w = t >> 2, jj = t & 3;
    float v = lds[w][jj];
    v += __shfl_xor(v, 4, 32);
    v += __shfl_xor(v, 8, 32);
    v += __shfl_xor(v, 16, 32);
    const float v0 = __shfl(v, 0, 32);   // reg sum
    const float v1 = __shfl(v, 1, 32);   // contain sum
    const float v2 = __shfl(v, 2, 32);   // noobj sum
    const float v3 = __shfl(v, 3, 32);   // cls sum
    if (t == 0) {
      const float invN = 1.0f / (float)N_BATCH;   // exact power of two
      const float reg = 5.0f * v0 * invN;
      const float con = v1 * invN;
      const float no  = 0.5f * v2 * invN;
      const float cl  = v3 * invN;
      out[0] = reg + con + no + cl;
      out[1] = reg;
      out[2] = con;
      out[3] = no;
      out[4] = cl;
    }
  }
}

extern "C" void kernel_launch(void* const* d_in, const int* in_sizes, int n_in,
                              void* d_out, int out_size, void* d_ws, size_t ws_size,
                              hipStream_t stream) {
  const float*         pred = (const float*)d_in[0];          // [N,S,S,30]
  const float*         tbox = (const float*)d_in[1];          // [N,S,S,4]
  const float*         tcls = (const float*)d_in[2];          // [N,S,S,20]
  const unsigned char* mask = (const unsigned char*)d_in[3];  // bool, 1 B/elem
  float* out     = (float*)d_out;                             // 5 floats
  float* partial = (float*)d_ws;                              // NBLOCKS*4 floats

  yolo_partial_kernel<<<NBLOCKS, NTHREADS, 0, stream>>>(pred, tbox, tcls, mask,
                                                        partial);
  yolo_final_kernel<<<1, 256, 0, stream>>>(partial, out);
}